// RandomBatchGeneralization_70111046139940
// MI455X (gfx1250) — compile-verified
//
#include <hip/hip_runtime.h>
#include <hip/hip_bf16.h>
#include <stdint.h>

// Problem constants (match the reference generator).
#define B_DIM 4096
#define D_DIM 4096
#define C_DIM 1000
#define NMAX  1024   // LDS staging capacity for index/mag arrays (N=409 fits)
#define MMAX  64     // per-row match list capacity

#define GLOBAL_AS __attribute__((address_space(1)))
#define LDS_AS    __attribute__((address_space(3)))

// ---- CDNA5 async global->LDS copy (probe-verified signature: AS1 int*, AS3 int*) ----
__device__ __forceinline__ void async_copy_b32(const void* g, void* l) {
#if defined(__gfx1250__) && __has_builtin(__builtin_amdgcn_global_load_async_to_lds_b32)
    void* gnc = const_cast<void*>(g);
    __builtin_amdgcn_global_load_async_to_lds_b32(
        (GLOBAL_AS int*)gnc, (LDS_AS int*)l, /*offset=*/0, /*cpol=*/0);
#else
    *(uint32_t*)l = *(const uint32_t*)g;
#endif
}

__device__ __forceinline__ void wait_async_copies() {
#if defined(__gfx1250__) && __has_builtin(__builtin_amdgcn_s_wait_asynccnt)
    __builtin_amdgcn_s_wait_asynccnt(0);
#elif defined(__gfx1250__) && __has_builtin(__builtin_amdgcn_global_load_async_to_lds_b32)
    asm volatile("s_wait_asynccnt 0x0" ::: "memory");
#endif
}

// ---- K1: last[r] = -1 ----
__global__ void init_last_kernel(int* __restrict__ last) {
    int r = blockIdx.x * blockDim.x + threadIdx.x;
    if (r < B_DIM) last[r] = -1;
}

// ---- K2: last[ref_index[i]] = max over i ----
__global__ void scatter_last_kernel(const int* __restrict__ ref,
                                    int* __restrict__ last, int N) {
    int i = blockIdx.x * blockDim.x + threadIdx.x;
    if (i < N) atomicMax(&last[ref[i]], i);
}

// ---- K3: ret[r] = (last<0) ? x[r] : x[ti]*m + x[r]*(1-m) ----
__global__ void xblend_kernel(const float* __restrict__ x,
                              const int*   __restrict__ tgt,
                              const float* __restrict__ mag,
                              const int*   __restrict__ last,
                              float* __restrict__ ret) {
    const int r  = blockIdx.y;
    const int c4 = blockIdx.x * blockDim.x + threadIdx.x;   // float4 column index
    const float4* xr   = (const float4*)(x   + (size_t)r * D_DIM);
    float4*       outp = (float4*)(ret + (size_t)r * D_DIM);
    const int li = last[r];
    if (li < 0) {
        outp[c4] = xr[c4];
        return;
    }
    const int   ti = tgt[li];
    const float m  = mag[li];
    const float om = 1.0f - m;
    const float4* xt = (const float4*)(x + (size_t)ti * D_DIM);
    __builtin_prefetch(&xt[c4], 0, 3);   // global_prefetch_b8
    float4 a = xr[c4];
    float4 b = xt[c4];
    float4 o;
    o.x = b.x * m + a.x * om;
    o.y = b.y * m + a.y * om;
    o.z = b.z * m + a.z * om;
    o.w = b.w * m + a.w * om;
    outp[c4] = o;
}

// ---- K4: full ret_y row computation, one block per row ----
// STAGED=true  : indices/mags staged to LDS via CDNA5 async copies, pure ds_* access
// STAGED=false : direct global access (N > NMAX fallback; never flat/generic mix)
template <bool STAGED>
__global__ void y_full_kernel(const float* __restrict__ y,
                              const int*   __restrict__ ref,
                              const int*   __restrict__ tgt,
                              const float* __restrict__ mag,
                              const int*   __restrict__ last,
                              float* __restrict__ ret_y, int N) {
    __shared__ int   s_tgt[NMAX];
    __shared__ int   s_ref[NMAX];
    __shared__ float s_mag[NMAX];
    __shared__ int   s_match[MMAX];
    __shared__ int   s_cnt;

    const int r    = blockIdx.x;
    const int nthr = blockDim.x;
    const int tid  = threadIdx.x;

    if (tid == 0) s_cnt = 0;
    if (STAGED) {
        for (int i = tid; i < N; i += nthr) {
            async_copy_b32(tgt + i, s_tgt + i);
            async_copy_b32(ref + i, s_ref + i);
            async_copy_b32(mag + i, s_mag + i);
        }
        wait_async_copies();
    }
    __syncthreads();

    const int last_r = last[r];

    // Parallel match-find: lanes scan strided j, append hits to LDS list.
    for (int j = tid; j < N; j += nthr) {
        const int tj = STAGED ? s_tgt[j] : tgt[j];
        if (tj == r && j > last_r) {
            int p = atomicAdd(&s_cnt, 1);     // ds_add_rtn_u32
            if (p < MMAX) s_match[p] = j;
        }
    }
    __syncthreads();

    const int  cnt   = s_cnt;
    const bool small = (cnt <= MMAX);
    if (small && cnt > 1 && tid == 0) {        // ascending j => reference fp order
        for (int a = 1; a < cnt; ++a) {
            int v = s_match[a], b = a - 1;
            while (b >= 0 && s_match[b] > v) { s_match[b + 1] = s_match[b]; --b; }
            s_match[b + 1] = v;
        }
    }
    __syncthreads();

    float wbase = 1.0f;
    if (last_r >= 0) {
        const float m  = mag[last_r];
        const float am = fabsf(m), a1 = fabsf(1.0f - m);
        wbase = a1 / (am + a1);
    }

    // Each thread owns columns c = tid, tid+nthr, ... (<=4 for 256 threads).
    float acc[4];
    int   cc[4];
    int   nk = 0;
    for (int c = tid; c < C_DIM; c += nthr) {
        cc[nk]  = c;
        acc[nk] = y[(size_t)r * C_DIM + c] * wbase;   // base = last set (or identity)
        ++nk;
    }

    if (small) {
        for (int q = 0; q < cnt; ++q) {
            const int   j  = s_match[q];
            const int   rj = STAGED ? s_ref[j] : ref[j];
            const float m  = STAGED ? s_mag[j] : mag[j];
            const float am = fabsf(m), a1 = fabsf(1.0f - m);
            const float s  = am / (am + a1);
            const float* yrow = y + (size_t)rj * C_DIM;
            for (int k = 0; k < nk; ++k) acc[k] += yrow[cc[k]] * s;
        }
    } else {   // overflow fallback: serial ascending scan (same fp order)
        for (int j = 0; j < N; ++j) {
            const int tj = STAGED ? s_tgt[j] : tgt[j];
            if (tj == r && j > last_r) {
                const int   rj = STAGED ? s_ref[j] : ref[j];
                const float m  = STAGED ? s_mag[j] : mag[j];
                const float am = fabsf(m), a1 = fabsf(1.0f - m);
                const float s  = am / (am + a1);
                const float* yrow = y + (size_t)rj * C_DIM;
                for (int k = 0; k < nk; ++k) acc[k] += yrow[cc[k]] * s;
            }
        }
    }

    for (int k = 0; k < nk; ++k)
        ret_y[(size_t)r * C_DIM + cc[k]] = acc[k];
}

extern "C" void kernel_launch(void* const* d_in, const int* in_sizes, int n_in,
                              void* d_out, int out_size, void* d_ws, size_t ws_size,
                              hipStream_t stream) {
    const float* x   = (const float*)d_in[0];
    const float* y   = (const float*)d_in[1];
    const int*   ref = (const int*)  d_in[2];
    const int*   tgt = (const int*)  d_in[3];
    const float* mag = (const float*)d_in[4];
    const int    N   = in_sizes[2];

    float* ret   = (float*)d_out;                       // B x D
    float* ret_y = ret + (size_t)B_DIM * D_DIM;         // B x C
    int*   last  = (int*)d_ws;                          // B ints of scratch

    init_last_kernel<<<(B_DIM + 255) / 256, 256, 0, stream>>>(last);
    scatter_last_kernel<<<(N + 255) / 256, 256, 0, stream>>>(ref, last, N);

    dim3 gx(D_DIM / (256 * 4), B_DIM, 1);               // float4-vectorized rows
    xblend_kernel<<<gx, 256, 0, stream>>>(x, tgt, mag, last, ret);

    if (N <= NMAX) {
        y_full_kernel<true><<<B_DIM, 256, 0, stream>>>(y, ref, tgt, mag, last, ret_y, N);
    } else {
        y_full_kernel<false><<<B_DIM, 256, 0, stream>>>(y, ref, tgt, mag, last, ret_y, N);
    }
}